// DynamicConvolution_42915313222210
// MI455X (gfx1250) — compile-verified
//
#include <hip/hip_runtime.h>
#include <cstdint>
#include <cstddef>

// Problem constants (from reference): x(B,L,D), depthwise causal conv K=7,
// mask = 0.5 ** boxfilter7(segment boundaries).
#define B_    4
#define L_    8192
#define D_    1024
#define K_    7
#define HALF_ 3

// Tiling: one block = 1 batch x TL l-positions x TD channels.
#define TL    32
#define TD    128
#define ROWS  (TL + K_ - 1)   // 38 x-rows staged (6 halo rows on the left)
#define NTHREADS 256          // 8 wave32

typedef unsigned int u32x4 __attribute__((ext_vector_type(4)));
typedef unsigned int u32x8 __attribute__((ext_vector_type(8)));

__global__ __launch_bounds__(NTHREADS)
void dwconv7_segmask_tdm_kernel(const float* __restrict__ x,
                                const float* __restrict__ bnd,
                                const float* __restrict__ w,      // (D,1,K): d*K + j
                                const float* __restrict__ bias,   // (D,)
                                float* __restrict__ out)
{
    __shared__ __align__(16) float tile[ROWS * TD];

    const int b  = blockIdx.z;
    const int l0 = blockIdx.y * TL;
    const int d0 = blockIdx.x * TD;
    const int t  = threadIdx.x;

    // ---- causal left padding: only blocks with l0 == 0 have rows < 0 ----
    const int pad       = (l0 == 0) ? (K_ - 1) : 0;   // 6 or 0
    const int row_start = l0 - (K_ - 1) + pad;        // first valid x row
    const int rows      = ROWS - pad;                 // rows the TDM moves

    if (pad != 0) {
        // zero-fill the 6 halo rows (DS stores; compiler waits DScnt at barrier)
        for (int i = t; i < pad * (TD / 4); i += NTHREADS) {
            reinterpret_cast<float4*>(tile)[i] = float4{0.f, 0.f, 0.f, 0.f};
        }
    }

    // ---------------- Tensor Data Mover: global -> LDS 2D tile ----------------
    // D# per CDNA5 ISA ch.8: group0 = {count/type, lds_addr, global_addr},
    // group1 = {mask/data_size, tensor dims, tile dims, strides}. 2D tile:
    // tile_dim0 = TD contiguous elements, tile_dim1 = rows, row stride = D.
    if (t == 0) {
        const uint64_t gaddr =
            (uint64_t)(uintptr_t)(x + (((size_t)b * L_ + (size_t)row_start) * D_ + d0));
        const uint32_t lds_addr =
            (uint32_t)(uintptr_t)(&tile[pad * TD]);   // low 32 bits = LDS byte offset

        const uint32_t tensor_d0 = (uint32_t)(D_ - d0);        // elems left in dim0
        const uint32_t tensor_d1 = (uint32_t)(L_ - row_start); // rows left in dim1
        const uint32_t tile_d0   = TD;                         // 128 elements
        const uint32_t tile_d1   = (uint32_t)rows;             // 32 or 38 rows
        const uint32_t stride0   = D_;                         // row stride (elems)

        u32x4 g0;
        g0.x = 1u;                                  // count=1, is_restore=0, gather=0
        g0.y = lds_addr;                            // lds_addr[31:0]
        g0.z = (uint32_t)(gaddr & 0xFFFFFFFFu);     // global_addr[31:0]
        g0.w = (uint32_t)((gaddr >> 32) & 0x01FFFFFFu) | (2u << 30); // addr[56:32], type=2

        u32x8 g1;
        g1.s0 = (2u << 16);                         // wg_mask=0, data_size=2 (4B), no opts
        g1.s1 = (tensor_d0 & 0xFFFFu) << 16;        // abar_addr=0 | tensor_dim0[15:0]
        g1.s2 = (tensor_d0 >> 16) | ((tensor_d1 & 0xFFFFu) << 16); // td0[31:16] | td1[15:0]
        g1.s3 = (tensor_d1 >> 16) | (tile_d0 << 16);               // td1[31:16] | tile_dim0
        g1.s4 = tile_d1;                            // tile_dim1 | tile_dim2=0 -> 2D
        g1.s5 = stride0;                            // tensor_dim0_stride[31:0]
        g1.s6 = 0u;                                 // stride0[47:32] | stride1[15:0]
        g1.s7 = 0u;                                 // stride1[47:16]

        const u32x4 gz = {0u, 0u, 0u, 0u};          // groups 2/3: unused (2D)

        asm volatile("tensor_load_to_lds %0, %1, %2, %3"
                     :
                     : "s"(g0), "s"(g1), "s"(gz), "s"(gz)
                     : "memory");
        __builtin_amdgcn_s_wait_tensorcnt(0);       // DMA landed in LDS
    }

    // ============ DMA-shadow work: nothing below touches the tile ============
    const int c4   = t & 31;     // channel group: 4 channels, float4 aligned
    const int lrow = t >> 5;     // 0..7, wave-uniform
    const int d    = d0 + c4 * 4;

    // Depthwise weights for this thread's 4 channels, all 7 taps -> registers.
    float4 w4[K_];
    #pragma unroll
    for (int j = 0; j < K_; ++j) {
        w4[j].x = w[(size_t)(d + 0) * K_ + j];
        w4[j].y = w[(size_t)(d + 1) * K_ + j];
        w4[j].z = w[(size_t)(d + 2) * K_ + j];
        w4[j].w = w[(size_t)(d + 3) * K_ + j];
    }
    const float4 bias4 = *reinterpret_cast<const float4*>(bias + d);
    const float* bnd_b = bnd + (size_t)b * L_;

    // mask[s] = 0.5 ** (# boundaries in [l-3, l+3] clipped to [0,L)) — tile-independent
    float mask[4];
    #pragma unroll
    for (int s = 0; s < 4; ++s) {
        const int l = l0 + lrow + s * 8;
        float cnt = 0.f;
        #pragma unroll
        for (int m = -HALF_; m <= HALF_; ++m) {
            const int lm = l + m;
            if (lm >= 0 && lm < L_) cnt += bnd_b[lm];   // wave-uniform address
        }
        mask[s] = exp2f(-cnt);
    }

    __syncthreads();             // tile (DMA + zero-fill) visible to all 8 waves

    // ---------------- compute: 4 l-positions per thread ----------------
    #pragma unroll
    for (int s = 0; s < 4; ++s) {
        const int lo = lrow + s * 8;     // 0..31 within tile
        const int l  = l0 + lo;

        float4 acc = bias4;
        #pragma unroll
        for (int j = 0; j < K_; ++j) {
            // y[l] = sum_j w[j] * x[l-6+j]; tile row r holds x row (l0-6+r),
            // so x[l-6+j] lives at tile row (lo + j).
            const float4 xv =
                *reinterpret_cast<const float4*>(&tile[(lo + j) * TD + c4 * 4]);
            acc.x += w4[j].x * xv.x;
            acc.y += w4[j].y * xv.y;
            acc.z += w4[j].z * xv.z;
            acc.w += w4[j].w * xv.w;
        }

        float4 o;
        o.x = acc.x * mask[s];
        o.y = acc.y * mask[s];
        o.z = acc.z * mask[s];
        o.w = acc.w * mask[s];
        *reinterpret_cast<float4*>(out + ((size_t)b * L_ + (size_t)l) * D_ + d) = o;
    }
}

extern "C" void kernel_launch(void* const* d_in, const int* in_sizes, int n_in,
                              void* d_out, int out_size, void* d_ws, size_t ws_size,
                              hipStream_t stream) {
    (void)in_sizes; (void)n_in; (void)d_ws; (void)ws_size; (void)out_size;
    const float* x    = (const float*)d_in[0];   // (B, L, D)
    const float* bnd  = (const float*)d_in[1];   // (B, L)
    const float* w    = (const float*)d_in[2];   // (D, 1, K)
    const float* bias = (const float*)d_in[3];   // (D,)
    float* out = (float*)d_out;                  // (B, L, D)

    dim3 grid(D_ / TD, L_ / TL, B_);             // (8, 256, 4)
    dwconv7_segmask_tdm_kernel<<<grid, NTHREADS, 0, stream>>>(x, bnd, w, bias, out);
}